// ConvFace_11441792876752
// MI455X (gfx1250) — compile-verified
//
#include <hip/hip_runtime.h>

// ---------------------------------------------------------------------------
// ConvFace attention pooling on MI455X (gfx1250, wave32, WMMA).
//
// Math restructure: softmax(s_j + const) == softmax(s_j), so the attention
// logit (Wk*xs+bk)·(Wq*xj+bq) reduces to (M^T xs + u)·xj with M = Wk^T Wq,
// u = Wq^T bk (precomputed once on device). bq drops out entirely.
// The two dense GEMMs (T = Xs*M, Y = Agg*Wc^T) use V_WMMA_F32_16X16X4_F32
// (full f32 precision, K=4 per instruction, 16 chained per 16x16 tile).
// ---------------------------------------------------------------------------

typedef __attribute__((ext_vector_type(2))) float v2f;
typedef __attribute__((ext_vector_type(8))) float v8f;

#define B_      4
#define C_      64
#define F_      20000
#define FP_     10000
#define KN      16        // neighbors
#define KJ      17        // self + neighbors
#define O_      128
#define TILE_F  16
#define NTILE   (FP_ / TILE_F)          // 625 tiles per batch
#define NBLK    (B_ * NTILE)            // 2500 workgroups
#define PITCH   65                      // LDS pitch for 64-wide rows (bank-conflict pad)
#define YPITCH  132                     // LDS pitch for 128-wide rows
#define INV_SQRT_DK 0.08838834764831845f
#define BN_EPS  1e-5f

// ---------------------------------------------------------------------------
// Kernel 0: M = Wk^T Wq (64x64), u = Wq^T bk (64), WcT (64x128 transpose).
// Tiny (0.5 MFLOP) one-block preamble.
// ---------------------------------------------------------------------------
__global__ __launch_bounds__(256) void convface_precompute(
    const float* __restrict__ Wk, const float* __restrict__ bk,
    const float* __restrict__ Wq, const float* __restrict__ Wc,
    float* __restrict__ Mg, float* __restrict__ ug, float* __restrict__ wct) {
  const int tid = threadIdx.x;
  for (int i = tid; i < 64 * 64; i += 256) {
    const int k = i >> 6, c = i & 63;
    float s = 0.f;
    #pragma unroll 8
    for (int d = 0; d < 128; ++d) s = fmaf(Wk[d * 64 + k], Wq[d * 64 + c], s);
    Mg[i] = s;
  }
  if (tid < 64) {
    float s = 0.f;
    #pragma unroll 8
    for (int d = 0; d < 128; ++d) s = fmaf(bk[d], Wq[d * 64 + tid], s);
    ug[tid] = s;
  }
  for (int i = tid; i < 64 * 128; i += 256) {
    const int k = i >> 7, o = i & 127;
    wct[i] = Wc[o * 64 + k];
  }
}

// ---------------------------------------------------------------------------
// Kernel 1: per-tile (16 faces) gather + attention + output GEMM.
// 256 threads = 8 wave32s. WMMA fragment layouts per CDNA5 ISA §7.12.2:
//   A (16x4 f32): lane m = lane&15, K = r + 2*(lane>>4)
//   B (4x16 f32): lane n = lane&15, K = r + 2*(lane>>4)
//   C/D (16x16):  row = v + 8*(lane>>4), col = lane&15
// ---------------------------------------------------------------------------
__global__ __launch_bounds__(256) void convface_main(
    const float* __restrict__ fea, const long long* __restrict__ ring_n,
    const long long* __restrict__ pool_idx, const float* __restrict__ bc,
    const float* __restrict__ Mg, const float* __restrict__ ug,
    const float* __restrict__ wct,
    float* __restrict__ ws_y, float* __restrict__ p1, float* __restrict__ p2) {
  __shared__ int   idx_s[TILE_F * KJ];
  __shared__ float xcat[TILE_F * KJ * PITCH];   // cat: self + 16 neighbors
  __shared__ float t_s[TILE_F * PITCH];         // T = Xs*M + u
  __shared__ float att_s[TILE_F * KJ];
  __shared__ float agg_s[TILE_F * PITCH];
  __shared__ float y_s[TILE_F * YPITCH];

  const int tid  = threadIdx.x;
  const int b    = blockIdx.x / NTILE;
  const int tile = blockIdx.x % NTILE;
  const int f0   = tile * TILE_F;
  const float* __restrict__ feab = fea + (size_t)b * C_ * F_;

  // --- stage gather indices (int64 -> int, values < 20000) ---
  for (int i = tid; i < TILE_F * KJ; i += 256) {
    const int f = i / KJ, j = i % KJ;
    long long idx = (j == 0) ? pool_idx[f0 + f]
                             : ring_n[((size_t)(b * FP_ + f0 + f)) * KN + (j - 1)];
    idx_s[i] = (int)idx;
  }
  __syncthreads();

  // --- gather cat[f][j][c] = fea[b][c][idx]; consecutive threads share a fea
  //     row (stride-F channel layout -> L2-resident 80KB rows) ---
  for (int i = tid; i < C_ * TILE_F * KJ; i += 256) {
    const int c = i / (TILE_F * KJ);
    const int r = i % (TILE_F * KJ);
    xcat[r * PITCH + c] = feab[(size_t)c * F_ + idx_s[r]];
  }
  __syncthreads();

  const int wave = tid >> 5;
  const int lane = tid & 31;
  const int half = lane >> 4;   // which 16-lane half
  const int ln   = lane & 15;   // A row / B-C-D column within tile

  // --- GEMM A: T[16x64] = Xs[16x64] * M[64x64] + u (waves 0..3, 16 cols each)
  if (wave < 4) {
    const int c0 = wave * 16;
    const float uval = ug[c0 + ln];
    v8f acc;
    #pragma unroll
    for (int v = 0; v < 8; ++v) acc[v] = uval;
    #pragma unroll
    for (int k0 = 0; k0 < 64; k0 += 4) {
      const int kk = k0 + 2 * half;
      v2f a, bb;
      a.x  = xcat[(ln * KJ + 0) * PITCH + kk];       // self row of cat
      a.y  = xcat[(ln * KJ + 0) * PITCH + kk + 1];
      bb.x = Mg[kk * 64 + c0 + ln];
      bb.y = Mg[(kk + 1) * 64 + c0 + ln];
      acc = __builtin_amdgcn_wmma_f32_16x16x4_f32(false, a, false, bb,
                                                  (short)0, acc, false, false);
    }
    #pragma unroll
    for (int v = 0; v < 8; ++v)
      t_s[(v + 8 * half) * PITCH + c0 + ln] = acc[v];
  }
  __syncthreads();

  // --- attention logits: s[f][j] = (T[f] . cat[f][j]) / sqrt(128) ---
  for (int i = tid; i < TILE_F * KJ; i += 256) {
    const int f = i / KJ;
    const float* tv = &t_s[f * PITCH];
    const float* xv = &xcat[i * PITCH];
    float s = 0.f;
    #pragma unroll 8
    for (int c = 0; c < C_; ++c) s = fmaf(tv[c], xv[c], s);
    att_s[i] = s * INV_SQRT_DK;
  }
  __syncthreads();

  // --- softmax over 17 (one thread per face) ---
  if (tid < TILE_F) {
    float m = -3.4e38f;
    #pragma unroll
    for (int j = 0; j < KJ; ++j) m = fmaxf(m, att_s[tid * KJ + j]);
    float e[KJ], sum = 0.f;
    #pragma unroll
    for (int j = 0; j < KJ; ++j) { e[j] = __expf(att_s[tid * KJ + j] - m); sum += e[j]; }
    const float inv = 1.f / sum;
    #pragma unroll
    for (int j = 0; j < KJ; ++j) att_s[tid * KJ + j] = e[j] * inv;
  }
  __syncthreads();

  // --- agg[f][c] = sum_j att[f][j] * cat[f][j][c] ---
  for (int i = tid; i < TILE_F * C_; i += 256) {
    const int f = i >> 6, c = i & 63;
    float s = 0.f;
    #pragma unroll
    for (int j = 0; j < KJ; ++j)
      s = fmaf(att_s[f * KJ + j], xcat[(f * KJ + j) * PITCH + c], s);
    agg_s[f * PITCH + c] = s;
  }
  __syncthreads();

  // --- GEMM B: Y[16x128] = Agg[16x64] * WcT[64x128] + bc (all 8 waves) ---
  {
    const int c0 = wave * 16;
    const float bcv = bc[c0 + ln];
    v8f acc;
    #pragma unroll
    for (int v = 0; v < 8; ++v) acc[v] = bcv;
    #pragma unroll
    for (int k0 = 0; k0 < 64; k0 += 4) {
      const int kk = k0 + 2 * half;
      v2f a, bb;
      a.x  = agg_s[ln * PITCH + kk];
      a.y  = agg_s[ln * PITCH + kk + 1];
      bb.x = wct[kk * O_ + c0 + ln];
      bb.y = wct[(kk + 1) * O_ + c0 + ln];
      acc = __builtin_amdgcn_wmma_f32_16x16x4_f32(false, a, false, bb,
                                                  (short)0, acc, false, false);
    }
    #pragma unroll
    for (int v = 0; v < 8; ++v)
      y_s[(v + 8 * half) * YPITCH + c0 + ln] = acc[v];
  }
  __syncthreads();

  // --- stage y to workspace ([B][O][Fp] layout) + per-tile BN partial sums ---
  for (int i = tid; i < TILE_F * O_; i += 256) {
    const int o = i >> 4, f = i & 15;
    ws_y[((size_t)(b * O_ + o)) * FP_ + f0 + f] = y_s[f * YPITCH + o];
  }
  if (tid < O_) {
    float s1 = 0.f, s2 = 0.f;
    #pragma unroll
    for (int f = 0; f < TILE_F; ++f) {
      const float v = y_s[f * YPITCH + tid];
      s1 += v;
      s2 = fmaf(v, v, s2);
    }
    p1[(size_t)blockIdx.x * O_ + tid] = s1;
    p2[(size_t)blockIdx.x * O_ + tid] = s2;
  }
}

// ---------------------------------------------------------------------------
// Kernel 2: deterministic BN stat reduction (fixed tile order per channel).
// ---------------------------------------------------------------------------
__global__ __launch_bounds__(128) void convface_bn_stats(
    const float* __restrict__ p1, const float* __restrict__ p2,
    const float* __restrict__ gamma, const float* __restrict__ beta,
    float* __restrict__ scale, float* __restrict__ shift) {
  const int o = threadIdx.x;
  if (o >= O_) return;
  float s1 = 0.f, s2 = 0.f;
  for (int t = 0; t < NBLK; ++t) {
    s1 += p1[(size_t)t * O_ + o];
    s2 += p2[(size_t)t * O_ + o];
  }
  const float n = (float)(B_ * FP_);
  const float mean = s1 / n;
  const float var  = s2 / n - mean * mean;
  const float sc   = gamma[o] * rsqrtf(var + BN_EPS);
  scale[o] = sc;
  shift[o] = beta[o] - mean * sc;
}

// ---------------------------------------------------------------------------
// Kernel 3: fused normalize + ReLU (streaming, HBM-bound).
// ---------------------------------------------------------------------------
__global__ __launch_bounds__(256) void convface_bn_apply(
    const float* __restrict__ ws_y, const float* __restrict__ scale,
    const float* __restrict__ shift, float* __restrict__ out) {
  const size_t i = (size_t)blockIdx.x * 256 + threadIdx.x;
  if (i >= (size_t)B_ * O_ * FP_) return;
  const int o = (int)((i / FP_) % O_);
  out[i] = fmaxf(fmaf(ws_y[i], scale[o], shift[o]), 0.f);
}

// ---------------------------------------------------------------------------
extern "C" void kernel_launch(void* const* d_in, const int* in_sizes, int n_in,
                              void* d_out, int out_size, void* d_ws, size_t ws_size,
                              hipStream_t stream) {
  (void)in_sizes; (void)n_in; (void)out_size;
  const float*     fea      = (const float*)d_in[0];
  const long long* ring_n   = (const long long*)d_in[1];
  const long long* pool_idx = (const long long*)d_in[2];
  // d_in[3] pos_embed: unused by the reference
  const float* Wk    = (const float*)d_in[4];
  const float* bk    = (const float*)d_in[5];
  const float* Wq    = (const float*)d_in[6];
  // d_in[7] bq: drops out (softmax shift invariance)
  const float* Wc    = (const float*)d_in[8];
  const float* bc    = (const float*)d_in[9];
  const float* gamma = (const float*)d_in[10];
  const float* beta  = (const float*)d_in[11];

  float* ws    = (float*)d_ws;
  float* Mg    = ws;             // 4096
  float* ug    = Mg + 4096;      // 64
  float* wct   = ug + 64;        // 8192
  float* scale = wct + 8192;     // 128
  float* shift = scale + 128;    // 128
  float* p1    = shift + 128;    // NBLK*128 = 320000
  float* p2    = p1 + (size_t)NBLK * O_;
  float* ws_y  = p2 + (size_t)NBLK * O_;   // B*O*Fp = 5,120,000
  const size_t need_bytes =
      (size_t)(4096 + 64 + 8192 + 128 + 128 + 2 * NBLK * O_ + B_ * O_ * FP_) * sizeof(float);
  if (ws_size < need_bytes) return;  // insufficient scratch; bail deterministically

  convface_precompute<<<1, 256, 0, stream>>>(Wk, bk, Wq, Wc, Mg, ug, wct);
  convface_main<<<NBLK, 256, 0, stream>>>(fea, ring_n, pool_idx, bc, Mg, ug, wct,
                                          ws_y, p1, p2);
  convface_bn_stats<<<1, 128, 0, stream>>>(p1, p2, gamma, beta, scale, shift);
  const int total = B_ * O_ * FP_;
  convface_bn_apply<<<(total + 255) / 256, 256, 0, stream>>>(ws_y, scale, shift,
                                                             (float*)d_out);
}